// NodeModel_ini_49503793053941
// MI455X (gfx1250) — compile-verified
//
#include <hip/hip_runtime.h>
#include <float.h>

typedef __attribute__((ext_vector_type(2))) float v2f;
typedef __attribute__((ext_vector_type(8))) float v8f;

#define N_NODES 100000
#define N_EDGES 1600000
#define EO      48           // EDGE_OUT
#define HID     128
#define NOUT    64
#define FEAT    (3*EO)       // 144
#define TILES   (N_NODES/16) // 6250 (N_NODES divisible by 16)
#define WAVES   4

// ---------------------------------------------------------------------------
// Phase 0: init workspace every call (harness does not re-poison between runs)
// ---------------------------------------------------------------------------
__global__ void gn_init_kernel(float* __restrict__ sum, float* __restrict__ mx,
                               float* __restrict__ cnt) {
    long long i = (long long)blockIdx.x * blockDim.x + threadIdx.x;
    long long tot = (long long)N_NODES * EO;
    if (i < tot) {
        sum[i] = 0.0f;
        mx[i]  = -FLT_MAX;   // sentinel for empty segments -> fixed to 0 later
    }
    if (i < N_NODES) cnt[i] = 0.0f;
}

// ---------------------------------------------------------------------------
// Phase 1: edge scatter (sum / max / count). One thread per (edge, feature)
// so edge_attr reads are fully coalesced (non-temporal: 307 MB streamed once);
// atomics land in L2 (accumulators are 2 x 19.2 MB << 192 MB L2).
// ---------------------------------------------------------------------------
__device__ __forceinline__ void atomicMaxF32(float* addr, float v) {
    // monotonic bit-pattern trick: one native integer atomic per call
    if (v >= 0.0f) atomicMax((int*)addr, __float_as_int(v));
    else           atomicMin((unsigned int*)addr, __float_as_uint(v));
}

__global__ void gn_scatter_kernel(const float* __restrict__ edge_attr,
                                  const int*   __restrict__ col,
                                  float* __restrict__ sum,
                                  float* __restrict__ mx,
                                  float* __restrict__ cnt) {
    long long t = (long long)blockIdx.x * blockDim.x + threadIdx.x;
    if (t >= (long long)N_EDGES * EO) return;
    int e = (int)(t / EO);
    int f = (int)(t % EO);
    int c = col[e];
    float v = __builtin_nontemporal_load(&edge_attr[t]);
    long long off = (long long)c * EO + f;
    atomicAdd(&sum[off], v);
    atomicMaxF32(&mx[off], v);
    if (f == 0) atomicAdd(&cnt[c], 1.0f);
}

// ---------------------------------------------------------------------------
// Phase 2: fused MLP with f32 WMMA (16x16x4). One wave32 per 16-node tile.
// All per-lane A-matrix data (24 x v2f) is preloaded so the GEMM loops are
// branch-free ds_load + v_wmma with overlapped load latency. Weights staged
// transposed in LDS; h tile spilled to a wave-private LDS region.
// ---------------------------------------------------------------------------
__global__ __launch_bounds__(WAVES * 32)
void gn_mlp_kernel(const float* __restrict__ sum, const float* __restrict__ mx,
                   const float* __restrict__ cnt,
                   const float* __restrict__ W1, const float* __restrict__ b1,
                   const float* __restrict__ W2, const float* __restrict__ b2,
                   float* __restrict__ out) {
    __shared__ float sW1t[FEAT * HID];    // sW1t[n*FEAT + k] = W1[k][n]
    __shared__ float sW2t[HID * NOUT];    // sW2t[n*HID  + k] = W2[k][n]
    __shared__ float sB1[HID];
    __shared__ float sB2[NOUT];
    __shared__ float sH[WAVES][16 * HID]; // per-wave h tile spill

    const int tid = threadIdx.x;

    for (int i = tid; i < FEAT * HID; i += WAVES * 32) {
        int k = i / HID, n = i % HID;
        sW1t[n * FEAT + k] = W1[i];
    }
    for (int i = tid; i < HID * NOUT; i += WAVES * 32) {
        int k = i / NOUT, n = i % NOUT;
        sW2t[n * HID + k] = W2[i];
    }
    for (int i = tid; i < HID;  i += WAVES * 32) sB1[i] = b1[i];
    for (int i = tid; i < NOUT; i += WAVES * 32) sB2[i] = b2[i];
    __syncthreads();   // last barrier: divergence below is wave-uniform only

    const int wave = tid >> 5;
    const int lane = tid & 31;
    const int tile = blockIdx.x * WAVES + wave;
    if (tile >= TILES) return;             // uniform per wave -> EXEC stays all-1

    const int m    = lane & 15;            // A row / B col within 16-wide tile
    const int grp  = lane >> 4;            // 0: K={0,1}, 1: K={2,3}
    const int node = tile * 16 + m;
    const long long base = (long long)node * EO;
    const float invc = 1.0f / fmaxf(cnt[node], 1.0f);

    // ---- preload all per-lane A fragments (loads issue back-to-back) ----
    v2f aS[12], aM[12];
    #pragma unroll
    for (int s = 0; s < 12; ++s)
        aS[s] = *(const v2f*)(sum + base + 4 * s + 2 * grp);
    #pragma unroll
    for (int s = 0; s < 12; ++s)
        aM[s] = *(const v2f*)(mx + base + 4 * s + 2 * grp);
    #pragma unroll
    for (int s = 0; s < 12; ++s) {         // empty-segment fixup: -FLT_MAX -> 0
        aM[s].x = (aM[s].x == -FLT_MAX) ? 0.0f : aM[s].x;
        aM[s].y = (aM[s].y == -FLT_MAX) ? 0.0f : aM[s].y;
    }

    // ---------------- GEMM1: feat[16x144] @ W1[144x128] ----------------
    v8f acc[8];
    #pragma unroll
    for (int j = 0; j < 8; ++j) acc[j] = (v8f){};

    #pragma unroll
    for (int s = 0; s < 12; ++s) {         // section 0: scatter-sum
        const int kk = 4 * s + 2 * grp;
        #pragma unroll
        for (int j = 0; j < 8; ++j) {
            v2f b = *(const v2f*)(&sW1t[(j * 16 + m) * FEAT + kk]);
            acc[j] = __builtin_amdgcn_wmma_f32_16x16x4_f32(
                false, aS[s], false, b, (short)0, acc[j], false, false);
        }
    }
    #pragma unroll
    for (int s = 0; s < 12; ++s) {         // section 1: scatter-max
        const int kk = EO + 4 * s + 2 * grp;
        #pragma unroll
        for (int j = 0; j < 8; ++j) {
            v2f b = *(const v2f*)(&sW1t[(j * 16 + m) * FEAT + kk]);
            acc[j] = __builtin_amdgcn_wmma_f32_16x16x4_f32(
                false, aM[s], false, b, (short)0, acc[j], false, false);
        }
    }
    #pragma unroll
    for (int s = 0; s < 12; ++s) {         // section 2: scatter-mean = sum*invc
        const int kk = 2 * EO + 4 * s + 2 * grp;
        v2f a;
        a.x = aS[s].x * invc;
        a.y = aS[s].y * invc;
        #pragma unroll
        for (int j = 0; j < 8; ++j) {
            v2f b = *(const v2f*)(&sW1t[(j * 16 + m) * FEAT + kk]);
            acc[j] = __builtin_amdgcn_wmma_f32_16x16x4_f32(
                false, a, false, b, (short)0, acc[j], false, false);
        }
    }

    // bias + ReLU, spill h tile to wave-private LDS (C layout: VGPR r ->
    // M = r + 8*grp, N = lane&15 within the 16-wide N tile)
    float* hrow = &sH[wave][0];
    #pragma unroll
    for (int j = 0; j < 8; ++j) {
        #pragma unroll
        for (int r = 0; r < 8; ++r) {
            int mm = grp * 8 + r;
            int n  = j * 16 + m;
            hrow[mm * HID + n] = fmaxf(acc[j][r] + sB1[n], 0.0f);
        }
    }

    // ---------------- GEMM2: h[16x128] @ W2[128x64] ----------------
    v8f acc2[4];
    #pragma unroll
    for (int j = 0; j < 4; ++j) acc2[j] = (v8f){};

    #pragma unroll
    for (int ks = 0; ks < HID / 4; ++ks) {       // 32 K-steps
        const int kk = ks * 4 + 2 * grp;
        v2f a = *(const v2f*)(&hrow[m * HID + kk]);
        #pragma unroll
        for (int j = 0; j < 4; ++j) {
            v2f b = *(const v2f*)(&sW2t[(j * 16 + m) * HID + kk]);
            acc2[j] = __builtin_amdgcn_wmma_f32_16x16x4_f32(
                false, a, false, b, (short)0, acc2[j], false, false);
        }
    }

    #pragma unroll
    for (int j = 0; j < 4; ++j) {
        #pragma unroll
        for (int r = 0; r < 8; ++r) {
            int mm = grp * 8 + r;
            int n  = j * 16 + m;
            __builtin_nontemporal_store(acc2[j][r] + sB2[n],
                &out[(long long)(tile * 16 + mm) * NOUT + n]);
        }
    }
}

// ---------------------------------------------------------------------------
extern "C" void kernel_launch(void* const* d_in, const int* in_sizes, int n_in,
                              void* d_out, int out_size, void* d_ws, size_t ws_size,
                              hipStream_t stream) {
    // inputs: x, edge_index, edge_attr, u, batch, W1, b1, W2, b2
    const int*   edge_index = (const int*)d_in[1];
    const float* edge_attr  = (const float*)d_in[2];
    const float* W1 = (const float*)d_in[5];
    const float* b1 = (const float*)d_in[6];
    const float* W2 = (const float*)d_in[7];
    const float* b2 = (const float*)d_in[8];
    float* out = (float*)d_out;

    float* sum = (float*)d_ws;                    // N_NODES*48 floats
    float* mx  = sum + (size_t)N_NODES * EO;      // N_NODES*48 floats
    float* cnt = mx  + (size_t)N_NODES * EO;      // N_NODES floats

    const int* col = edge_index + N_EDGES;        // edge_index[1][:]

    {
        long long tot = (long long)N_NODES * EO;
        int blocks = (int)((tot + 255) / 256);
        gn_init_kernel<<<blocks, 256, 0, stream>>>(sum, mx, cnt);
    }
    {
        long long tot = (long long)N_EDGES * EO;
        int blocks = (int)((tot + 255) / 256);
        gn_scatter_kernel<<<blocks, 256, 0, stream>>>(edge_attr, col, sum, mx, cnt);
    }
    {
        int blocks = (TILES + WAVES - 1) / WAVES;
        gn_mlp_kernel<<<blocks, WAVES * 32, 0, stream>>>(sum, mx, cnt,
                                                         W1, b1, W2, b2, out);
    }
}